// GAT_36713380446846
// MI455X (gfx1250) — compile-verified
//
#include <hip/hip_runtime.h>
#include <hip/hip_bf16.h>
#include <math.h>

// ---------------- problem constants (match reference) ----------------
#define GN    16          // graphs
#define NPG   512         // nodes per graph
#define NN    (GN*NPG)    // 8192 nodes
#define FIN   128
#define HH_   8           // heads
#define CC    128         // channels per head
#define HC    1024        // HH_*CC

#define KC    64          // GEMM K-chunk staged in LDS
#define WLD   68          // LDS row stride (floats): 272B = 16B-aligned, 4-bank rotate

typedef float v2f __attribute__((ext_vector_type(2)));
typedef float v8f __attribute__((ext_vector_type(8)));

// =====================================================================
// CSR construction over dst (deterministic: int atomics + canonical sort)
// =====================================================================
__global__ void k_zero_i32(int* p, int n) {
    int i = blockIdx.x * blockDim.x + threadIdx.x;
    if (i < n) p[i] = 0;
}

__global__ void k_count_deg(const int* __restrict__ dst, int E, int Etot,
                            int* __restrict__ deg) {
    int e = blockIdx.x * blockDim.x + threadIdx.x;
    if (e >= Etot) return;
    int d = (e < E) ? dst[e] : (e - E);   // self-loop tail
    atomicAdd(&deg[d], 1);
}

// single-block exclusive scan of deg[0..NN) -> rowptr[0..NN], 1024 threads x 8
__global__ void k_scan(const int* __restrict__ deg, int* __restrict__ rowptr) {
    __shared__ int sdata[1024];
    int t = threadIdx.x;
    int base = t * 8;
    int loc[8];
    int s = 0;
    #pragma unroll
    for (int i = 0; i < 8; ++i) { loc[i] = s; s += deg[base + i]; }
    sdata[t] = s;
    __syncthreads();
    for (int off = 1; off < 1024; off <<= 1) {
        int v = (t >= off) ? sdata[t - off] : 0;
        __syncthreads();
        sdata[t] += v;
        __syncthreads();
    }
    int excl = sdata[t] - s;
    #pragma unroll
    for (int i = 0; i < 8; ++i) rowptr[base + i] = excl + loc[i];
    if (t == 1023) rowptr[NN] = sdata[1023];
}

__global__ void k_fill(const int* __restrict__ dst, int E, int Etot,
                       const int* __restrict__ rowptr, int* __restrict__ cursor,
                       int* __restrict__ ebuf) {
    int e = blockIdx.x * blockDim.x + threadIdx.x;
    if (e >= Etot) return;
    int d = (e < E) ? dst[e] : (e - E);
    int pos = rowptr[d] + atomicAdd(&cursor[d], 1);
    ebuf[pos] = e;
}

// canonicalize bucket order (insertion sort ascending) => determinism
__global__ void k_sort_buckets(const int* __restrict__ rowptr, int* __restrict__ ebuf) {
    int n = blockIdx.x * blockDim.x + threadIdx.x;
    if (n >= NN) return;
    int r0 = rowptr[n], r1 = rowptr[n + 1];
    for (int i = r0 + 1; i < r1; ++i) {
        int key = ebuf[i];
        int j = i - 1;
        while (j >= r0 && ebuf[j] > key) { ebuf[j + 1] = ebuf[j]; --j; }
        ebuf[j + 1] = key;
    }
}

// =====================================================================
// fp32 WMMA GEMM with async-LDS staging of the W strip.
//   out[nrows,cout] = X[nrows,K] @ W[cout,K]^T  (+bias,relu)
// Block = 256 threads = 8 waves = 128x16 output tile (8 m-tiles x 1 n-tile).
// The shared 16-row W strip is staged in LDS in K-chunks of 64 via
// global_load_async_to_lds_b128 (ASYNCcnt), double-buffered so the copy of
// chunk c+1 overlaps WMMA on chunk c. A fragments stream from global (L2).
//   A 16x4 layout: lanes 0-15 rows M=0..15 hold K=k,k+1 ; lanes 16-31 K=k+2,k+3
//   C/D layout:    VGPR j -> (M=j, N=lane) lanes 0-15 ; (M=j+8, N=lane-16)
// =====================================================================
__global__ void k_gemm_wmma(const float* __restrict__ X, const float* __restrict__ W,
                            const float* __restrict__ bias, float* __restrict__ out,
                            int nrows, int K, int cout, int fuse_bias_relu) {
    __shared__ float wbuf[2][16 * WLD];

    int t     = threadIdx.x;
    int wave  = t >> 5;
    int lane  = t & 31;
    int tilesN = cout >> 4;
    int bm = blockIdx.x / tilesN;
    int bn = blockIdx.x % tilesN;
    int m0 = bm * 128 + wave * 16;
    int n0 = bn * 16;

    int lr    = lane & 15;
    int khalf = (lane >> 4) << 1;                 // 0 or 2
    const float* xrow = X + (size_t)(m0 + lr) * K;

    // per-thread slice of the 16x64 W strip copy: 16B each, 256 threads = 4KB
    int cr  = t >> 4;                             // strip row 0..15
    int cc4 = (t & 15) << 2;                      // col 0,4,...,60
    const float* gsrc = W + (size_t)(n0 + cr) * K + cc4;

    auto issue_copy = [&](int ch, int b) {
        unsigned      ldsa = (unsigned)(uintptr_t)(&wbuf[b][cr * WLD + cc4]);
        unsigned long long ga = (unsigned long long)(uintptr_t)(gsrc + (size_t)ch * KC);
        asm volatile("global_load_async_to_lds_b128 %0, %1, off"
                     :: "v"(ldsa), "v"(ga) : "memory");
    };

    int nch = K / KC;
    issue_copy(0, 0);

    v8f acc = {};
    for (int ch = 0; ch < nch; ++ch) {
        if (ch + 1 < nch) {
            issue_copy(ch + 1, (ch + 1) & 1);
            asm volatile("s_wait_asynccnt 0x1" ::: "memory");
        } else {
            asm volatile("s_wait_asynccnt 0x0" ::: "memory");
        }
        __syncthreads();                           // strip published to all waves

        const float* wl = &wbuf[ch & 1][lr * WLD + khalf];
        const float* xk = xrow + ch * KC + khalf;
        #pragma unroll
        for (int kk = 0; kk < KC; kk += 4) {
            v2f a = *reinterpret_cast<const v2f*>(xk + kk);
            v2f b = *reinterpret_cast<const v2f*>(wl + kk);
            acc = __builtin_amdgcn_wmma_f32_16x16x4_f32(
                /*neg_a=*/false, a, /*neg_b=*/false, b,
                /*c_mod=*/(short)0, acc, /*reuse_a=*/false, /*reuse_b=*/false);
        }
        __syncthreads();                           // buffer free before next overwrite
    }

    int mo  = (lane < 16) ? 0 : 8;
    int col = n0 + lr;
    float bv = fuse_bias_relu ? bias[col] : 0.0f;
    #pragma unroll
    for (int j = 0; j < 8; ++j) {
        float v = acc[j];
        if (fuse_bias_relu) v = fmaxf(v + bv, 0.0f);
        out[(size_t)(m0 + j + mo) * cout + col] = v;
    }
}

// =====================================================================
// attention logit coefficients: al_s[n,h] = sum_c h[n,h,c]*a_src[h,c]
// one wave per (n,h); 128 channels striped 4-per-lane; xor-tree reduce
// =====================================================================
__global__ void k_att_dots(const float* __restrict__ hbuf,
                           const float* __restrict__ a_src,
                           const float* __restrict__ a_dst,
                           float* __restrict__ als, float* __restrict__ ald) {
    int gw   = (blockIdx.x * blockDim.x + threadIdx.x) >> 5;
    int lane = threadIdx.x & 31;
    int n = gw >> 3, h = gw & 7;
    if (n >= NN) return;
    const float* hr = hbuf + (size_t)n * HC + h * CC;
    const float* as = a_src + h * CC;
    const float* ad = a_dst + h * CC;
    float ps = 0.f, pd = 0.f;
    #pragma unroll
    for (int j = 0; j < 4; ++j) {
        float v = hr[lane + 32 * j];
        ps += v * as[lane + 32 * j];
        pd += v * ad[lane + 32 * j];
    }
    #pragma unroll
    for (int off = 16; off > 0; off >>= 1) {
        ps += __shfl_xor(ps, off, 32);
        pd += __shfl_xor(pd, off, 32);
    }
    if (lane == 0) { als[n * HH_ + h] = ps; ald[n * HH_ + h] = pd; }
}

// =====================================================================
// gather-based scatter-softmax + aggregation (+bias+relu)
// one block per dst node, one wave per head; channels across lanes
// =====================================================================
__global__ void k_gat_agg(const float* __restrict__ hbuf,
                          const float* __restrict__ als, const float* __restrict__ ald,
                          const int* __restrict__ rowptr, const int* __restrict__ ebuf,
                          const int* __restrict__ srcArr, int E,
                          const float* __restrict__ bias, float* __restrict__ out) {
    int n    = blockIdx.x;
    int h    = threadIdx.x >> 5;
    int lane = threadIdx.x & 31;
    int r0 = rowptr[n], r1 = rowptr[n + 1];
    float aldv = ald[n * HH_ + h];

    // pass 1: segment max of leaky_relu logits
    float m = -INFINITY;
    for (int i = r0; i < r1; ++i) {
        int e = ebuf[i];
        int s = (e < E) ? srcArr[e] : (e - E);
        float x = als[s * HH_ + h] + aldv;
        x = (x > 0.f) ? x : 0.2f * x;
        m = fmaxf(m, x);
    }
    // pass 2: softmax-weighted sum of source features
    float z = 0.f, a0 = 0.f, a1 = 0.f, a2 = 0.f, a3 = 0.f;
    for (int i = r0; i < r1; ++i) {
        int e = ebuf[i];
        int s = (e < E) ? srcArr[e] : (e - E);
        float x = als[s * HH_ + h] + aldv;
        x = (x > 0.f) ? x : 0.2f * x;
        float p = expf(x - m);
        z += p;
        const float* hr = hbuf + (size_t)s * HC + h * CC;
        a0 += p * hr[lane];
        a1 += p * hr[lane + 32];
        a2 += p * hr[lane + 64];
        a3 += p * hr[lane + 96];
    }
    float inv = 1.0f / (z + 1e-16f);
    int cb = h * CC + lane;
    float* orow = out + (size_t)n * HC;
    orow[cb]      = fmaxf(a0 * inv + bias[cb],      0.f);
    orow[cb + 32] = fmaxf(a1 * inv + bias[cb + 32], 0.f);
    orow[cb + 64] = fmaxf(a2 * inv + bias[cb + 64], 0.f);
    orow[cb + 96] = fmaxf(a3 * inv + bias[cb + 96], 0.f);
}

// =====================================================================
// row-wise L2 normalize in place (block per row)
// =====================================================================
__global__ void k_rownorm(float* __restrict__ x) {
    __shared__ float red[256];
    int n = blockIdx.x, t = threadIdx.x;
    float* row = x + (size_t)n * HC;
    float ss = 0.f;
    #pragma unroll
    for (int j = 0; j < 4; ++j) { float v = row[t + 256 * j]; ss += v * v; }
    red[t] = ss;
    __syncthreads();
    for (int off = 128; off > 0; off >>= 1) {
        if (t < off) red[t] += red[t + off];
        __syncthreads();
    }
    float scale = 1.0f / fmaxf(sqrtf(red[0]), 1e-12f);
    #pragma unroll
    for (int j = 0; j < 4; ++j) row[t + 256 * j] *= scale;
}

// =====================================================================
// small MLP layer: out[N,32] = relu(X[N,32] @ W[32,32]^T + b), W staged in LDS
// =====================================================================
__global__ void k_small_mlp(const float* __restrict__ X, const float* __restrict__ W,
                            const float* __restrict__ b, float* __restrict__ out) {
    __shared__ float sw[32 * 32];
    int t = threadIdx.x;
    #pragma unroll
    for (int j = 0; j < 4; ++j) sw[t + 256 * j] = W[t + 256 * j];
    __syncthreads();
    int o = t & 31;
    int row = blockIdx.x * 8 + (t >> 5);
    const float* xr = X + row * 32;
    float acc = b[o];
    #pragma unroll
    for (int k = 0; k < 32; ++k) acc += xr[k] * sw[o * 32 + k];
    out[row * 32 + o] = fmaxf(acc, 0.f);
}

// =====================================================================
// per-graph H^T H (32x32 over 512 nodes) via fp32 WMMA; 4 tiles per graph
// HH[g, d*32+e] = sum_n g3[gbase+n, d] * g3[gbase+n, e]
// =====================================================================
__global__ void k_hth_wmma(const float* __restrict__ g3, float* __restrict__ HHout) {
    int ww   = (blockIdx.x * blockDim.x + threadIdx.x) >> 5;  // 64 waves total
    int lane = threadIdx.x & 31;
    int g    = ww >> 2;
    int tile = ww & 3;
    if (g >= GN) return;
    int m0 = (tile >> 1) << 4;     // 0 or 16 (d)
    int n0 = (tile & 1) << 4;      // 0 or 16 (e)
    int lr    = lane & 15;
    int khalf = (lane >> 4) << 1;
    const float* base = g3 + (size_t)g * NPG * 32;

    v8f acc = {};
    for (int k = 0; k < NPG; k += 4) {
        v2f a, b;
        a.x = base[(k + khalf)     * 32 + m0 + lr];   // A[d, n] = g3[n, d]
        a.y = base[(k + khalf + 1) * 32 + m0 + lr];
        b.x = base[(k + khalf)     * 32 + n0 + lr];   // B[n, e] = g3[n, e]
        b.y = base[(k + khalf + 1) * 32 + n0 + lr];
        acc = __builtin_amdgcn_wmma_f32_16x16x4_f32(
            false, a, false, b, (short)0, acc, false, false);
    }
    int mo = (lane < 16) ? 0 : 8;
    #pragma unroll
    for (int j = 0; j < 8; ++j)
        HHout[(size_t)g * 1024 + (m0 + j + mo) * 32 + (n0 + lr)] = acc[j];
}

// =====================================================================
// final graph MLP (fused, one block of 512 threads): HH[16,1024] -> o[16,2]
// =====================================================================
__global__ void k_final_mlp(const float* __restrict__ HHin,
                            const float* __restrict__ Wm1, const float* __restrict__ bm1,
                            const float* __restrict__ Wm2, const float* __restrict__ bm2,
                            const float* __restrict__ Wm3, const float* __restrict__ bm3,
                            float* __restrict__ outp) {
    __shared__ float s1[GN * 32];
    __shared__ float s2[GN * 32];
    int t = threadIdx.x;
    int g = t >> 5, o = t & 31;
    // stage 1: [16,1024] @ [1024,32]^T
    {
        const float* hr = HHin + (size_t)g * 1024;
        const float* wr = Wm1 + (size_t)o * 1024;
        float acc = bm1[o];
        for (int k = 0; k < 1024; ++k) acc += hr[k] * wr[k];
        s1[g * 32 + o] = fmaxf(acc, 0.f);
    }
    __syncthreads();
    // stage 2: [16,32] @ [32,32]^T
    {
        float acc = bm2[o];
        #pragma unroll
        for (int k = 0; k < 32; ++k) acc += s1[g * 32 + k] * Wm2[o * 32 + k];
        s2[g * 32 + o] = fmaxf(acc, 0.f);
    }
    __syncthreads();
    // stage 3: [16,32] @ [32,2]^T
    if (t < GN * 2) {
        int gg = t >> 1, oo = t & 1;
        float acc = bm3[oo];
        #pragma unroll
        for (int k = 0; k < 32; ++k) acc += s2[gg * 32 + k] * Wm3[oo * 32 + k];
        outp[gg * 2 + oo] = fmaxf(acc, 0.f);
    }
}

// =====================================================================
// host-side orchestration
// =====================================================================
extern "C" void kernel_launch(void* const* d_in, const int* in_sizes, int n_in,
                              void* d_out, int out_size, void* d_ws, size_t ws_size,
                              hipStream_t stream) {
    const float* x    = (const float*)d_in[0];
    const int*   ei   = (const int*)  d_in[1];
    const float* W1   = (const float*)d_in[2];
    const float* as1  = (const float*)d_in[3];
    const float* ad1  = (const float*)d_in[4];
    const float* b1   = (const float*)d_in[5];
    const float* W2   = (const float*)d_in[6];
    const float* as2  = (const float*)d_in[7];
    const float* ad2  = (const float*)d_in[8];
    const float* b2   = (const float*)d_in[9];
    const float* Ws1  = (const float*)d_in[10];
    const float* bs1  = (const float*)d_in[11];
    const float* Ws2  = (const float*)d_in[12];
    const float* bs2  = (const float*)d_in[13];
    const float* Ws3  = (const float*)d_in[14];
    const float* bs3  = (const float*)d_in[15];
    const float* Wm1  = (const float*)d_in[16];
    const float* bm1  = (const float*)d_in[17];
    const float* Wm2  = (const float*)d_in[18];
    const float* bm2  = (const float*)d_in[19];
    const float* Wm3  = (const float*)d_in[20];
    const float* bm3  = (const float*)d_in[21];

    const int E    = in_sizes[1] / 2;
    const int Etot = E + NN;
    const int* srcArr = ei;
    const int* dstArr = ei + E;

    // ---- workspace carve (256B aligned) ----
    size_t off = 0;
    auto carve = [&](size_t bytes) -> char* {
        char* p = (char*)d_ws + off;
        off += (bytes + 255) & ~(size_t)255;
        return p;
    };
    float* hbuf   = (float*)carve((size_t)NN * HC * sizeof(float));  // linear proj
    float* hout   = (float*)carve((size_t)NN * HC * sizeof(float));  // layer outputs
    float* als    = (float*)carve((size_t)NN * HH_ * sizeof(float));
    float* ald    = (float*)carve((size_t)NN * HH_ * sizeof(float));
    float* g1     = (float*)carve((size_t)NN * 32 * sizeof(float));
    float* g2     = (float*)carve((size_t)NN * 32 * sizeof(float));
    float* g3     = (float*)carve((size_t)NN * 32 * sizeof(float));
    int*   deg    = (int*)  carve((size_t)NN * sizeof(int));
    int*   cursor = (int*)  carve((size_t)NN * sizeof(int));
    int*   rowptr = (int*)  carve((size_t)(NN + 1) * sizeof(int));
    int*   ebuf   = (int*)  carve((size_t)Etot * sizeof(int));

    float* HHout = (float*)d_out;                 // [16,1024]
    float* oOut  = (float*)d_out + GN * 1024;     // [16,2]

    // ---- CSR build (once; shared by both layers) ----
    k_zero_i32<<<(2 * NN + 255) / 256, 256, 0, stream>>>(deg, 2 * NN); // deg+cursor contiguous
    k_count_deg<<<(Etot + 255) / 256, 256, 0, stream>>>(dstArr, E, Etot, deg);
    k_scan<<<1, 1024, 0, stream>>>(deg, rowptr);
    k_fill<<<(Etot + 255) / 256, 256, 0, stream>>>(dstArr, E, Etot, rowptr, cursor, ebuf);
    k_sort_buckets<<<NN / 256, 256, 0, stream>>>(rowptr, ebuf);

    // ---- GAT layer 1 ----
    k_gemm_wmma<<<(NN / 128) * (HC / 16), 256, 0, stream>>>(x, W1, nullptr, hbuf,
                                                            NN, FIN, HC, 0);
    k_att_dots<<<(NN * HH_) / 8, 256, 0, stream>>>(hbuf, as1, ad1, als, ald);
    k_gat_agg<<<NN, 256, 0, stream>>>(hbuf, als, ald, rowptr, ebuf, srcArr, E, b1, hout);

    // ---- GAT layer 2 ----
    k_gemm_wmma<<<(NN / 128) * (HC / 16), 256, 0, stream>>>(hout, W2, nullptr, hbuf,
                                                            NN, HC, HC, 0);
    k_att_dots<<<(NN * HH_) / 8, 256, 0, stream>>>(hbuf, as2, ad2, als, ald);
    k_gat_agg<<<NN, 256, 0, stream>>>(hbuf, als, ald, rowptr, ebuf, srcArr, E, b2, hout);

    // ---- normalize + node MLP ----
    k_rownorm<<<NN, 256, 0, stream>>>(hout);
    k_gemm_wmma<<<(NN / 128) * (32 / 16), 256, 0, stream>>>(hout, Ws1, bs1, g1,
                                                            NN, HC, 32, 1);
    k_small_mlp<<<NN / 8, 256, 0, stream>>>(g1, Ws2, bs2, g2);
    k_small_mlp<<<NN / 8, 256, 0, stream>>>(g2, Ws3, bs3, g3);

    // ---- per-graph H^T H + readout MLP ----
    k_hth_wmma<<<8, 256, 0, stream>>>(g3, HHout);   // 64 waves = 16 graphs x 4 tiles
    k_final_mlp<<<1, 512, 0, stream>>>(HHout, Wm1, bm1, Wm2, bm2, Wm3, bm3, oOut);
}